// PRGNN_55087250538649
// MI455X (gfx1250) — compile-verified
//
#include <hip/hip_runtime.h>

#define NNODES 16384
#define NEDGES 262144
#define SDIM   8
#define FDIM   32
#define WCOLS  320   // 256 (Wk as 8x 32x32) + 32 (bk) + 32 (root)

typedef __attribute__((ext_vector_type(2))) float v2f;
typedef __attribute__((ext_vector_type(8))) float v8f;

// ---------------------------------------------------------------------------
// Zero agg (N*32) and pooled (32) at start of each launch.
__global__ void ecc_zero(float* __restrict__ agg, float* __restrict__ pooled) {
    int i = blockIdx.x * blockDim.x + threadIdx.x;
    if (i < NNODES * FDIM) agg[i] = 0.0f;
    if (i < FDIM) pooled[i] = 0.0f;
}

// ---------------------------------------------------------------------------
// Build W' (32 x 320): cols [0,256) = Wk[k][i,o] -> W'[i, k*32+o]
//                      cols [256,288) = bk[i,o], cols [288,320) = root[i,o]
__global__ void ecc_prep(const float* __restrict__ Wk, const float* __restrict__ bk,
                         const float* __restrict__ root, float* __restrict__ Wp) {
    int i = blockIdx.x * blockDim.x + threadIdx.x;
    if (i >= FDIM * WCOLS) return;
    int r = i / WCOLS, c = i % WCOLS;
    float v;
    if (c < 256)      { int k = c >> 5, o = c & 31; v = Wk[k * (FDIM * FDIM) + r * FDIM + o]; }
    else if (c < 288) { int o = c - 256;            v = bk[r * FDIM + o]; }
    else              { int o = c - 288;            v = root[r * FDIM + o]; }
    Wp[i] = v;
}

// ---------------------------------------------------------------------------
// U(N x 320) = A(N x 32, row stride lda) @ B(32 x 320), fp32 WMMA 16x16x4.
// One wave per 16x16 C tile; K=32 -> 8 chained v_wmma_f32_16x16x4_f32.
// A frag (16x4 f32): lanes 0-15 -> M=lane, VGPR0=K k, VGPR1=K k+1;
//                    lanes 16-31 -> M=lane-16, K k+2 / k+3.  (ISA 7.12.2)
// C/D (16x16 f32): VGPR v -> M = v + 8*(lane>=16), N = lane&15.
__global__ void ecc_gemm_wmma(const float* __restrict__ A, int lda,
                              const float* __restrict__ B, float* __restrict__ U) {
    const int NT = WCOLS / 16;                       // 20 col tiles
    int wave = (blockIdx.x * blockDim.x + threadIdx.x) >> 5;
    int lane = threadIdx.x & 31;
    int row0 = (wave / NT) * 16;
    int col0 = (wave % NT) * 16;
    if (row0 >= NNODES) return;
    int hi = lane >> 4;                              // 0 | 1
    int lo = lane & 15;
    v8f c = {};
#pragma unroll
    for (int k = 0; k < FDIM; k += 4) {
        int kk = k + hi * 2;
        const float* arow = A + (size_t)(row0 + lo) * lda + kk;
        const float* bcol = B + (size_t)kk * WCOLS + col0 + lo;
        v2f a, b;
        a.x = arow[0];
        a.y = arow[1];
        b.x = bcol[0];
        b.y = bcol[WCOLS];
        c = __builtin_amdgcn_wmma_f32_16x16x4_f32(false, a, false, b,
                                                  (short)0, c, false, false);
    }
    float* urow = U + (size_t)(row0 + hi * 8) * WCOLS + col0 + lo;
#pragma unroll
    for (int v = 0; v < 8; ++v)
        urow[(size_t)v * WCOLS] = c[v];
}

// ---------------------------------------------------------------------------
// One wave32 per edge: lane o computes msg[o] = sum_k e[k]*U[src][k*32+o]
//                                              + U[src][256+o], atomic-add to agg[tgt].
__global__ void ecc_edge(const int* __restrict__ src, const int* __restrict__ tgt,
                         const float* __restrict__ e, const float* __restrict__ U,
                         float* __restrict__ agg) {
    int wave = (blockIdx.x * blockDim.x + threadIdx.x) >> 5;
    int lane = threadIdx.x & 31;
    if (wave >= NEDGES) return;
    int s = src[wave];
    int t = tgt[wave];
    const float* Ur = U + (size_t)s * WCOLS;
    float ev[SDIM];
#pragma unroll
    for (int k = 0; k < SDIM; ++k) ev[k] = e[(size_t)wave * SDIM + k];
    float m = Ur[256 + lane];                        // bk contribution
#pragma unroll
    for (int k = 0; k < SDIM; ++k) m += ev[k] * Ur[k * FDIM + lane];
    atomicAdd(&agg[(size_t)t * FDIM + lane], m);
}

// ---------------------------------------------------------------------------
// h' = relu(agg + U[:,288:320] + b); also reset agg to 0 for the next layer.
__global__ void ecc_update(const float* __restrict__ U, float* __restrict__ agg,
                           const float* __restrict__ bias, float* __restrict__ hout) {
    int i = blockIdx.x * blockDim.x + threadIdx.x;
    if (i >= NNODES * FDIM) return;
    int n = i >> 5, o = i & 31;
    float v = agg[i] + U[(size_t)n * WCOLS + 288 + o] + bias[o];
    hout[i] = v > 0.0f ? v : 0.0f;
    agg[i] = 0.0f;
}

// ---------------------------------------------------------------------------
// pooled[o] = sum_n h2[n,o]  (block-local LDS reduction, then 32 atomics/block)
__global__ void ecc_pool(const float* __restrict__ h2, float* __restrict__ pooled) {
    __shared__ float s[FDIM];
    int t = threadIdx.x;
    if (t < FDIM) s[t] = 0.0f;
    __syncthreads();
    int lane = t & 31;
    int rpb  = blockDim.x >> 5;                      // rows handled per block pass
    float acc = 0.0f;
    for (int row = blockIdx.x * rpb + (t >> 5); row < NNODES; row += gridDim.x * rpb)
        acc += h2[(size_t)row * FDIM + lane];
    atomicAdd(&s[lane], acc);
    __syncthreads();
    if (t < FDIM) atomicAdd(&pooled[t], s[t]);
}

// ---------------------------------------------------------------------------
// out = relu(dot(pooled, Wd) + bd)  — single wave32, shuffle reduction.
__global__ void ecc_final(const float* __restrict__ pooled, const float* __restrict__ Wd,
                          const float* __restrict__ bd, float* __restrict__ out) {
    int lane = threadIdx.x;
    float v = pooled[lane] * Wd[lane];
#pragma unroll
    for (int off = 16; off > 0; off >>= 1) v += __shfl_down(v, off, 32);
    if (lane == 0) {
        float r = v + bd[0];
        out[0] = r > 0.0f ? r : 0.0f;
    }
}

// ---------------------------------------------------------------------------
extern "C" void kernel_launch(void* const* d_in, const int* in_sizes, int n_in,
                              void* d_out, int out_size, void* d_ws, size_t ws_size,
                              hipStream_t stream) {
    const float* x     = (const float*)d_in[0];    // (N, 33), col 32 == 1.0 (unused)
    const int*   src   = (const int*)  d_in[1];
    const int*   tgt   = (const int*)  d_in[2];
    const float* e     = (const float*)d_in[3];
    const float* Wk1   = (const float*)d_in[4];
    const float* bk1   = (const float*)d_in[5];
    const float* root1 = (const float*)d_in[6];
    const float* b1    = (const float*)d_in[7];
    const float* Wk2   = (const float*)d_in[8];
    const float* bk2   = (const float*)d_in[9];
    const float* root2 = (const float*)d_in[10];
    const float* b2    = (const float*)d_in[11];
    const float* Wd    = (const float*)d_in[12];
    const float* bd    = (const float*)d_in[13];
    float* out = (float*)d_out;

    // Workspace layout (floats): ~27.4 MB total
    float* W1p    = (float*)d_ws;                        // 32*320   = 10240
    float* W2p    = W1p + FDIM * WCOLS;                  // 10240
    float* U      = W2p + FDIM * WCOLS;                  // N*320    = 5242880
    float* agg    = U   + (size_t)NNODES * WCOLS;        // N*32     = 524288
    float* h1     = agg + (size_t)NNODES * FDIM;         // 524288
    float* h2     = h1  + (size_t)NNODES * FDIM;         // 524288
    float* pooled = h2  + (size_t)NNODES * FDIM;         // 32

    const int TB = 256;
    const int zeroBlocks = (NNODES * FDIM + TB - 1) / TB;            // 2048
    const int prepBlocks = (FDIM * WCOLS + TB - 1) / TB;             // 40
    const int gemmBlocks = (NNODES / 16) * (WCOLS / 16) / (TB / 32); // 2560
    const int edgeBlocks = NEDGES / (TB / 32);                       // 32768
    const int updBlocks  = (NNODES * FDIM + TB - 1) / TB;            // 2048

    ecc_zero<<<zeroBlocks, TB, 0, stream>>>(agg, pooled);
    ecc_prep<<<prepBlocks, TB, 0, stream>>>(Wk1, bk1, root1, W1p);
    ecc_prep<<<prepBlocks, TB, 0, stream>>>(Wk2, bk2, root2, W2p);

    // Layer 1: A = x with row stride 33 (drops the constant last column)
    ecc_gemm_wmma<<<gemmBlocks, TB, 0, stream>>>(x, 33, W1p, U);
    ecc_edge<<<edgeBlocks, TB, 0, stream>>>(src, tgt, e, U, agg);
    ecc_update<<<updBlocks, TB, 0, stream>>>(U, agg, b1, h1);

    // Layer 2
    ecc_gemm_wmma<<<gemmBlocks, TB, 0, stream>>>(h1, FDIM, W2p, U);
    ecc_edge<<<edgeBlocks, TB, 0, stream>>>(src, tgt, e, U, agg);
    ecc_update<<<updBlocks, TB, 0, stream>>>(U, agg, b2, h2);

    // Pool + readout
    ecc_pool<<<128, TB, 0, stream>>>(h2, pooled);
    ecc_final<<<1, 32, 0, stream>>>(pooled, Wd, bd, out);
}